// GCN_Net_39238821216832
// MI455X (gfx1250) — compile-verified
//
#include <hip/hip_runtime.h>

typedef __attribute__((ext_vector_type(2))) float v2f;
typedef __attribute__((ext_vector_type(8))) float v8f;

#define F_IN  256
#define F_HID 128
#define F_OUT 32

// ---------------- degree / normalization ----------------

__global__ void k_init_deg(float* deg, int n) {
  int i = blockIdx.x * blockDim.x + threadIdx.x;
  if (i < n) deg[i] = 1.0f;  // self-loop contributes 1 to every degree
}

__global__ void k_count_deg(const long long* __restrict__ ei, float* deg, int E) {
  int e = blockIdx.x * blockDim.x + threadIdx.x;
  if (e < E) atomicAdd(&deg[(int)ei[(long long)E + e]], 1.0f);
}

__global__ void k_dinv(float* deg, int n) {
  int i = blockIdx.x * blockDim.x + threadIdx.x;
  if (i < n) deg[i] = rsqrtf(deg[i]);  // deg >= 1 always (self-loop)
}

__global__ void k_norm(const long long* __restrict__ ei, const float* __restrict__ dinv,
                       float* __restrict__ norm, int E) {
  int e = blockIdx.x * blockDim.x + threadIdx.x;
  if (e < E) norm[e] = dinv[(int)ei[e]] * dinv[(int)ei[(long long)E + e]];
}

// ---------------- f32 WMMA GEMM with fused prologue/epilogue ----------------
// H[M x NCOL]   = act(A) @ B        (act = optional bias+relu applied while staging A)
// AGG[M x NCOL] = H * dinv[row]^2   (self-loop init of the scatter accumulator)
// One wave = one 16x16 C tile via V_WMMA_F32_16X16X4_F32; A tile staged in LDS
// with float4 loads and shared by all waves of the block.
// Requires M % (16*MT) == 0 so EXEC stays all-ones around every WMMA.

template <int K, int NCOL, int MT, bool RELU>
__global__ void k_gemm_wmma(const float* __restrict__ A, const float* __restrict__ B,
                            const float* __restrict__ bias, const float* __restrict__ dinv,
                            float* __restrict__ H, float* __restrict__ AGG) {
  constexpr int WAVES_N = NCOL / 16;
  const int wave = threadIdx.x >> 5;
  const int lane = threadIdx.x & 31;
  const int l16  = lane & 15;
  const int hi   = lane >> 4;  // lane half selects K pair (ISA 32-bit A/B layout)
  const int mt   = wave / WAVES_N;
  const int nc   = (wave % WAVES_N) * 16;

  __shared__ float As[MT * 16 * K];
  const long long arow0 = (long long)blockIdx.x * (MT * 16);

  // stage A tile (optionally fused bias+relu), 16B-vectorized
  constexpr int ELEMS4 = MT * 16 * K / 4;
  for (int i = threadIdx.x; i < ELEMS4; i += blockDim.x) {
    int g = i * 4;
    int r = g / K;
    int c = g - r * K;
    float4 v = *(const float4*)(A + (arow0 + r) * K + c);
    if constexpr (RELU) {
      float4 bb = *(const float4*)(bias + c);
      v.x = fmaxf(v.x + bb.x, 0.0f);
      v.y = fmaxf(v.y + bb.y, 0.0f);
      v.z = fmaxf(v.z + bb.z, 0.0f);
      v.w = fmaxf(v.w + bb.w, 0.0f);
    }
    *(float4*)(&As[g]) = v;
  }
  __syncthreads();

  const float* as = &As[(mt * 16 + l16) * K];
  v8f acc = {};
#pragma unroll 4
  for (int k = 0; k < K; k += 4) {
    v2f a, b;
    // A 16x4 (32-bit): VGPR0 = K {k | k+2}, VGPR1 = K {k+1 | k+3} per lane half
    a.x = as[k + 2 * hi + 0];
    a.y = as[k + 2 * hi + 1];
    // B 4x16 (row striped across lanes): VGPR r holds row k+r (lanes 0-15) / k+r+2 (16-31)
    const float* bp = &B[(long long)(k + 2 * hi) * NCOL + nc + l16];
    b.x = bp[0];
    b.y = bp[NCOL];
    acc = __builtin_amdgcn_wmma_f32_16x16x4_f32(false, a, false, b, (short)0, acc,
                                                false, false);
  }

  // C 16x16 f32: VGPR r -> row r (lanes 0-15) / row r+8 (lanes 16-31), col = lane%16
  const long long rbase = arow0 + mt * 16;
  float* hp = H + rbase * NCOL + nc;
  float* gp = AGG + rbase * NCOL + nc;
#pragma unroll
  for (int r = 0; r < 8; ++r) {
    int rr = r + 8 * hi;
    float di = dinv[(int)(rbase + rr)];
    float c = acc[r];
    hp[(long long)rr * NCOL + l16] = c;
    gp[(long long)rr * NCOL + l16] = c * di * di;  // fused self-loop init
  }
}

// ---------------- edge scatter (one thread per edge x 4 features) ----------------

template <int F>
__global__ void k_edge_scatter(const float* __restrict__ h, const long long* __restrict__ ei,
                               const float* __restrict__ norm, float* __restrict__ agg, int E) {
  constexpr int Q = F / 4;
  long long t = (long long)blockIdx.x * blockDim.x + threadIdx.x;
  if (t < (long long)E * Q) {
    int e = (int)(t / Q);
    int q = (int)(t - (long long)e * Q) * 4;
    int s = (int)ei[e];
    int d = (int)ei[(long long)E + e];
    float w = norm[e];
    float4 v = *(const float4*)(h + (long long)s * F + q);
    float* a = agg + (long long)d * F + q;
    atomicAdd(a + 0, v.x * w);
    atomicAdd(a + 1, v.y * w);
    atomicAdd(a + 2, v.z * w);
    atomicAdd(a + 3, v.w * w);
  }
}

// ---------------- row log-softmax (F_OUT == warpSize == 32) ----------------

__global__ void k_logsoftmax32(float* __restrict__ out, const float* __restrict__ b, int n) {
  int w = (int)(((long long)blockIdx.x * blockDim.x + threadIdx.x) >> 5);
  int lane = threadIdx.x & 31;
  if (w >= n) return;
  long long idx = (long long)w * 32 + lane;
  float v = out[idx] + b[lane];
  float mx = v;
#pragma unroll
  for (int o = 16; o > 0; o >>= 1) mx = fmaxf(mx, __shfl_xor(mx, o, 32));
  float ex = expf(v - mx);
  float s = ex;
#pragma unroll
  for (int o = 16; o > 0; o >>= 1) s += __shfl_xor(s, o, 32);
  out[idx] = v - mx - logf(s);
}

// ---------------- launch ----------------

static inline int cdiv(long long a, long long b) { return (int)((a + b - 1) / b); }

extern "C" void kernel_launch(void* const* d_in, const int* in_sizes, int n_in,
                              void* d_out, int out_size, void* d_ws, size_t ws_size,
                              hipStream_t stream) {
  const float*     x  = (const float*)d_in[0];
  const long long* ei = (const long long*)d_in[1];  // int64 per reference
  const float*     W1 = (const float*)d_in[2];
  const float*     b1 = (const float*)d_in[3];
  const float*     W2 = (const float*)d_in[4];
  const float*     b2 = (const float*)d_in[5];
  float*           out = (float*)d_out;

  const int N = in_sizes[0] / F_IN;  // 100000
  const int E = in_sizes[1] / 2;     // 1600000

  // workspace (floats): dinv[N] | norm[E] | h1[N*128] | agg1[N*128] | h2[N*32]
  float* dinv = (float*)d_ws;
  float* norm = dinv + N;
  float* h1   = norm + E;
  float* agg1 = h1 + (long long)N * F_HID;
  float* h2   = agg1 + (long long)N * F_HID;

  // normalization: deg (with self-loops) -> dinv -> per-edge norm
  k_init_deg<<<cdiv(N, 256), 256, 0, stream>>>(dinv, N);
  k_count_deg<<<cdiv(E, 256), 256, 0, stream>>>(ei, dinv, E);
  k_dinv<<<cdiv(N, 256), 256, 0, stream>>>(dinv, N);
  k_norm<<<cdiv(E, 256), 256, 0, stream>>>(ei, dinv, norm, E);

  // layer 1: h1 = x @ W1; agg1 = h1 * dinv^2 (fused) ; then edge scatter
  k_gemm_wmma<F_IN, F_HID, 1, false><<<N / 16, (F_HID / 16) * 32, 0, stream>>>(
      x, W1, nullptr, dinv, h1, agg1);
  k_edge_scatter<F_HID><<<cdiv((long long)E * (F_HID / 4), 256), 256, 0, stream>>>(
      h1, ei, norm, agg1, E);

  // layer 2: h2 = relu(agg1 + b1) @ W2 (bias+relu fused into A staging);
  //          out = h2 * dinv^2 (fused) ; then edge scatter into d_out
  k_gemm_wmma<F_HID, F_OUT, 2, true><<<N / 32, 2 * (F_OUT / 16) * 32, 0, stream>>>(
      agg1, W2, b1, dinv, h2, out);
  k_edge_scatter<F_OUT><<<cdiv((long long)E * (F_OUT / 4), 256), 256, 0, stream>>>(
      h2, ei, norm, out, E);

  // bias + log_softmax, in place on d_out (one wave per row of 32)
  k_logsoftmax32<<<cdiv(N, 8), 256, 0, stream>>>(out, b2, N);
}